// S4Layer_71270687309834
// MI455X (gfx1250) — compile-verified
//
#include <hip/hip_runtime.h>

typedef __attribute__((ext_vector_type(2))) float v2f;
typedef __attribute__((ext_vector_type(8))) float v8f;
typedef __attribute__((ext_vector_type(4))) unsigned int u32x4;
typedef __attribute__((ext_vector_type(8))) int i32x8;
typedef __attribute__((ext_vector_type(4))) int i32x4;

#define BATCH    8
#define D_MODEL  512
#define STATE_N  64
#define SEQ_L    2048
#define CHUNK_T  64
#define NCHUNK   (SEQ_L / CHUNK_T)      // 32
#define NCOL     (BATCH * NCHUNK)       // 256 columns per channel; col = c*8 + b
#define LDA      68                     // LDS row pitch (64 + 4 pad)

#if __has_builtin(__builtin_amdgcn_tensor_load_to_lds)
#define HAVE_TDM 1
#endif

// ---------------------------------------------------------------------------
// Shared WMMA pass: acc{0,1} += A(64x64, LDS) @ B(64x64, LDS) for this wave's
// two 16x16 output tiles (mi, ni0) and (mi, ni0+1). 32 wmma per call.
// A-frag: lanes 0-15 K={k,k+1}, lanes 16-31 K={k+2,k+3}; B mirrored.
// ---------------------------------------------------------------------------
__device__ __forceinline__ void mm_pass(const float (*Ab)[LDA], const float (*Bb)[LDA],
                                        int mi, int ni0, int lane,
                                        v8f& acc0, v8f& acc1)
{
    const int m  = lane & 15;
    const int kb = (lane < 16) ? 0 : 2;
    #pragma unroll
    for (int k = 0; k < 64; k += 4) {
        v2f a, b0, b1;
        a.x  = Ab[mi * 16 + m][k + kb];
        a.y  = Ab[mi * 16 + m][k + kb + 1];
        b0.x = Bb[k + kb][ni0 * 16 + m];
        b0.y = Bb[k + kb + 1][ni0 * 16 + m];
        b1.x = Bb[k + kb][(ni0 + 1) * 16 + m];
        b1.y = Bb[k + kb + 1][(ni0 + 1) * 16 + m];
        acc0 = __builtin_amdgcn_wmma_f32_16x16x4_f32(false, a, false, b0, (short)0, acc0, false, false);
        acc1 = __builtin_amdgcn_wmma_f32_16x16x4_f32(false, a, false, b1, (short)0, acc1, false, false);
    }
}

// ---------------------------------------------------------------------------
// Kernel U: per-channel chunk input projection  U = P_d @ X_d  (complex).
// grid (4 col-tiles, 512 channels), 256 threads. P regenerated in LDS.
// ---------------------------------------------------------------------------
__global__ __launch_bounds__(256) void ssm_chunk_proj_kernel(
    const float* __restrict__ x,
    const float* __restrict__ log_A_real,
    const float* __restrict__ A_imag,
    const float* __restrict__ B_real,
    const float* __restrict__ B_imag,
    const float* __restrict__ log_dt,
    float* __restrict__ Ur, float* __restrict__ Ui)
{
    __shared__ float Ab[64][LDA];
    __shared__ float Bb[64][LDA];

    const int ct  = blockIdx.x;        // 64-column tile
    const int d   = blockIdx.y;
    const int tid = threadIdx.x;
    const int wave = tid >> 5, lane = tid & 31;
    const int mi = wave & 3, ni0 = (wave >> 2) * 2;

    // stage X tile: X[j][cv] = x[b, d, c*64 + j], col = ct*64+cv, c=col>>3, b=col&7
    #pragma unroll
    for (int i = 0; i < 16; ++i) {
        const int idx = tid + i * 256;
        const int j = idx & 63, cv = idx >> 6;
        const int col = ct * 64 + cv;
        const int b = col & 7, c = col >> 3;
        Bb[j][cv] = x[((size_t)b * D_MODEL + d) * SEQ_L + c * CHUNK_T + j];
    }

    float wr = 0.f, wi = 0.f, br = 0.f, bi = 0.f;
    if (tid < 64) {
        const int pidx = d * STATE_N + tid;
        const float dt = __expf(log_dt[d]);
        const float ar = -__expf(log_A_real[pidx]);
        const float ai = A_imag[pidx];
        const float mg = __expf(ar * dt);
        wr = mg * __cosf(ai * dt);  wi = mg * __sinf(ai * dt);
        br = B_real[pidx];          bi = B_imag[pidx];
        // phase-1 A-matrix: Pr[n][63-k] = Re(B * w^k)
        float pr = br, pi = bi;
        for (int k = 0; k < 64; ++k) {
            Ab[tid][63 - k] = pr;
            const float nr = pr * wr - pi * wi;
            pi = pr * wi + pi * wr;  pr = nr;
        }
    }
    __syncthreads();

    v8f acc0 = {0,0,0,0,0,0,0,0}, acc1 = {0,0,0,0,0,0,0,0};
    mm_pass(Ab, Bb, mi, ni0, lane, acc0, acc1);
    __syncthreads();

    if (tid < 64) {                    // phase-2 A-matrix: Pi[n][63-k] = Im(B*w^k)
        float pr = br, pi = bi;
        for (int k = 0; k < 64; ++k) {
            Ab[tid][63 - k] = pi;
            const float nr = pr * wr - pi * wi;
            pi = pr * wi + pi * wr;  pr = nr;
        }
    }
    __syncthreads();

    v8f acc2 = {0,0,0,0,0,0,0,0}, acc3 = {0,0,0,0,0,0,0,0};
    mm_pass(Ab, Bb, mi, ni0, lane, acc2, acc3);

    const int ncol = lane & 15, mbase = (lane < 16) ? 0 : 8;
    #pragma unroll
    for (int i = 0; i < 8; ++i) {
        const int n = mi * 16 + mbase + i;
        const int col0 = ct * 64 + ni0 * 16 + ncol;
        const size_t base = ((size_t)d * STATE_N + n) * NCOL;
        Ur[base + col0]      = acc0[i];
        Ur[base + col0 + 16] = acc1[i];
        Ui[base + col0]      = acc2[i];
        Ui[base + col0 + 16] = acc3[i];
    }
}

// ---------------------------------------------------------------------------
// Kernel S: 32-step chunk-state recurrence, in-place U -> S_prev.
//   s_c = w^64 * s_{c-1} + u_c ;  store S_prev[col=c*8+b] = s_{c-1}.
// 262144 independent threads (d, n, b); only sequential part of the layer.
// ---------------------------------------------------------------------------
__global__ __launch_bounds__(256) void ssm_chunk_scan_kernel(
    const float* __restrict__ log_A_real,
    const float* __restrict__ A_imag,
    const float* __restrict__ log_dt,
    float* __restrict__ Ur, float* __restrict__ Ui)
{
    const int t = blockIdx.x * 256 + threadIdx.x;     // 0 .. 262143
    const int b = t & 7, n = (t >> 3) & 63, d = t >> 9;
    const int pidx = d * STATE_N + n;

    const float dt = __expf(log_dt[d]);
    const float ar = -__expf(log_A_real[pidx]);
    const float ai = A_imag[pidx];
    const float mg = __expf(ar * dt);
    float wr = mg * __cosf(ai * dt), wi = mg * __sinf(ai * dt);
    #pragma unroll
    for (int q = 0; q < 6; ++q) {                     // w -> w^64
        const float nr = wr * wr - wi * wi;
        wi = 2.f * wr * wi;  wr = nr;
    }

    float sr = 0.f, si = 0.f;
    const size_t base = (size_t)pidx * NCOL;
    for (int c = 0; c < NCHUNK; ++c) {
        const size_t idx = base + (size_t)c * 8 + b;
        const float ur = Ur[idx], ui = Ui[idx];
        Ur[idx] = sr;  Ui[idx] = si;                  // S_prev
        const float nr = fmaf(wr, sr, fmaf(-wi, si, ur));
        si = fmaf(wr, si, fmaf(wi, sr, ui));
        sr = nr;
    }
}

// ---------------------------------------------------------------------------
// Kernel Y: y = L_d @ X + Er @ Sr - Ei @ Si + D*x   (3 WMMA passes, one acc).
// L_d built from Toeplitz row K (LDS ds_add_f32 reduction over states).
// ---------------------------------------------------------------------------
__global__ __launch_bounds__(256) void ssm_chunk_out_kernel(
    const float* __restrict__ x,
    const float* __restrict__ log_A_real,
    const float* __restrict__ A_imag,
    const float* __restrict__ B_real,
    const float* __restrict__ B_imag,
    const float* __restrict__ C_real,
    const float* __restrict__ C_imag,
    const float* __restrict__ Dvec,
    const float* __restrict__ log_dt,
    const float* __restrict__ Sr,     // Ur after scan
    const float* __restrict__ Si,     // Ui after scan
    float* __restrict__ y)
{
    __shared__ float Ab[64][LDA];
    __shared__ float Bb[64][LDA];
    __shared__ float Krow[64];

    const int ct  = blockIdx.x;
    const int d   = blockIdx.y;
    const int tid = threadIdx.x;
    const int wave = tid >> 5, lane = tid & 31;
    const int mi = wave & 3, ni0 = (wave >> 2) * 2;

    float wr = 0.f, wi = 0.f, cbr = 0.f, cbi = 0.f, cdr = 0.f, cdi = 0.f;
    if (tid < 64) {
        const int pidx = d * STATE_N + tid;
        const float dt = __expf(log_dt[d]);
        const float ar = -__expf(log_A_real[pidx]);
        const float ai = A_imag[pidx];
        const float mg = __expf(ar * dt);
        wr = mg * __cosf(ai * dt);  wi = mg * __sinf(ai * dt);
        const float br = B_real[pidx], bi = B_imag[pidx];
        cdr = C_real[pidx] * dt;  cdi = C_imag[pidx] * dt;      // C*dt
        cbr = cdr * br - cdi * bi;  cbi = cdr * bi + cdi * br;  // (C*dt)*B
        Krow[tid] = 0.f;
    }
    __syncthreads();
    if (tid < 64) {                   // K[k] = sum_n Re((C*dt*B)_n * w_n^k)
        float gr = cbr, gi = cbi;
        for (int k = 0; k < 64; ++k) {
            atomicAdd(&Krow[k], gr);  // LDS ds_add_f32
            const float nr = gr * wr - gi * wi;
            gi = gr * wi + gi * wr;  gr = nr;
        }
    }
    __syncthreads();

    // pass 1: A = lower-triangular Toeplitz L[t][j] = K[t-j]; B = X tile
    #pragma unroll
    for (int i = 0; i < 16; ++i) {
        const int idx = tid + i * 256;
        const int tt = idx >> 6, j = idx & 63;
        Ab[tt][j] = (tt >= j) ? Krow[tt - j] : 0.f;
    }
    #pragma unroll
    for (int i = 0; i < 16; ++i) {
        const int idx = tid + i * 256;
        const int j = idx & 63, cv = idx >> 6;
        const int col = ct * 64 + cv;
        const int b = col & 7, c = col >> 3;
        Bb[j][cv] = x[((size_t)b * D_MODEL + d) * SEQ_L + c * CHUNK_T + j];
    }
    __syncthreads();
    v8f acc0 = {0,0,0,0,0,0,0,0}, acc1 = {0,0,0,0,0,0,0,0};
    mm_pass(Ab, Bb, mi, ni0, lane, acc0, acc1);
    __syncthreads();

    // pass 2: A = Er[t][n] = Re(C*dt*w^(t+1)); B = Sr tile
    if (tid < 64) {
        float gr = cdr * wr - cdi * wi, gi = cdr * wi + cdi * wr;   // (C*dt)*w
        for (int tt = 0; tt < 64; ++tt) {
            Ab[tt][tid] = gr;
            const float nr = gr * wr - gi * wi;
            gi = gr * wi + gi * wr;  gr = nr;
        }
    }
    #pragma unroll
    for (int i = 0; i < 16; ++i) {
        const int idx = tid + i * 256;
        const int n = idx >> 6, cv = idx & 63;
        Bb[n][cv] = Sr[((size_t)d * STATE_N + n) * NCOL + ct * 64 + cv];
    }
    __syncthreads();
    mm_pass(Ab, Bb, mi, ni0, lane, acc0, acc1);
    __syncthreads();

    // pass 3: A = -Ei[t][n] = -Im(C*dt*w^(t+1)); B = Si tile
    if (tid < 64) {
        float gr = cdr * wr - cdi * wi, gi = cdr * wi + cdi * wr;
        for (int tt = 0; tt < 64; ++tt) {
            Ab[tt][tid] = -gi;
            const float nr = gr * wr - gi * wi;
            gi = gr * wi + gi * wr;  gr = nr;
        }
    }
    #pragma unroll
    for (int i = 0; i < 16; ++i) {
        const int idx = tid + i * 256;
        const int n = idx >> 6, cv = idx & 63;
        Bb[n][cv] = Si[((size_t)d * STATE_N + n) * NCOL + ct * 64 + cv];
    }
    __syncthreads();
    mm_pass(Ab, Bb, mi, ni0, lane, acc0, acc1);

    // epilogue: y[b,d,c*64+t] = acc + D*x ; per-lane 8 consecutive l values
    const float Dd = Dvec[d];
    const int ncol = lane & 15, mbase = (lane < 16) ? 0 : 8;
    {
        const int col0 = ct * 64 + ni0 * 16 + ncol;
        const int b0 = col0 & 7, c0 = col0 >> 3;
        const size_t base0 = ((size_t)b0 * D_MODEL + d) * SEQ_L + c0 * CHUNK_T + mi * 16 + mbase;
        const int col1 = col0 + 16;
        const int b1 = col1 & 7, c1 = col1 >> 3;
        const size_t base1 = ((size_t)b1 * D_MODEL + d) * SEQ_L + c1 * CHUNK_T + mi * 16 + mbase;
        #pragma unroll
        for (int i = 0; i < 8; ++i) {
            y[base0 + i] = fmaf(Dd, x[base0 + i], acc0[i]);
            y[base1 + i] = fmaf(Dd, x[base1 + i], acc1[i]);
        }
    }
}

#ifdef HAVE_TDM
// ---------------------------------------------------------------------------
// TDM descriptor build + issue: 2-D tile of 4-byte elements, global -> LDS,
// with DMA-inserted row padding (32 DWORD rows -> pitch 33 in LDS).
// D# layout per CDNA5 ISA.
// 6-arg builtin form (clang-23 / therock-10.0 headers).
// ---------------------------------------------------------------------------
__device__ __forceinline__ void tdm_load_2d(unsigned lds_off, const void* gaddr,
                                            unsigned tile_w, unsigned tile_h,
                                            unsigned tensor_w, unsigned tensor_h,
                                            unsigned row_stride)
{
    const unsigned long long ga = (unsigned long long)(uintptr_t)gaddr;
    u32x4 g0;
    g0[0] = 1u;                                              // count=1, user D#
    g0[1] = lds_off;                                         // lds_addr
    g0[2] = (unsigned)(ga & 0xffffffffu);                    // global_addr lo
    g0[3] = (unsigned)((ga >> 32) & 0x01ffffffu) | 0x80000000u; // addr hi | type=2
    i32x8 g1;
    g1[0] = (int)((2u << 16) | (1u << 20) | (4u << 22));     // 4B elems; pad_en; every 32 dwords +1 dword
    g1[1] = (int)((tensor_w & 0xffffu) << 16);               // tensor_dim0 lo16
    g1[2] = (int)(((tensor_w >> 16) & 0xffffu) | ((tensor_h & 0xffffu) << 16));
    g1[3] = (int)(((tensor_h >> 16) & 0xffffu) | ((tile_w & 0xffffu) << 16));
    g1[4] = (int)(tile_h & 0xffffu);                         // tile_dim1 (tile_dim2=0)
    g1[5] = (int)row_stride;                                 // tensor_dim0_stride lo32
    g1[6] = 0;
    g1[7] = 0;
    const i32x4 gz4 = {0, 0, 0, 0};                          // 2-D tensor: groups 2/3 unused
    const i32x8 gz8 = {0, 0, 0, 0, 0, 0, 0, 0};
    __builtin_amdgcn_tensor_load_to_lds(g0, g1, gz4, gz4, gz8, 0);
}
#endif

// ---------------------------------------------------------------------------
// Kernel O: out[b,o,l] = sum_c W[o,c]*y[b,c,l] + b_out[o] + x[b,o,l]
// TDM variant: wave 0 double-buffers W(64x32)/Y(32x32) K-stage tiles into LDS
// via tensor_load_to_lds (TENSORcnt-ordered), all 8 waves run WMMA.
// ---------------------------------------------------------------------------
__global__ __launch_bounds__(256) void outproj_wmma_kernel(
    const float* __restrict__ y,
    const float* __restrict__ x,
    const float* __restrict__ W,
    const float* __restrict__ b_out,
    float* __restrict__ out)
{
    const int b  = blockIdx.z;
    const int o0 = blockIdx.y * 64;
    const int l0 = blockIdx.x * 32;

    const int tid  = threadIdx.x;
    const int wave = tid >> 5;
    const int lane = tid & 31;
    const int oo = (wave & 3) * 16;
    const int ll = (wave >> 2) * 16;

    const float* Wp = W + (size_t)o0 * D_MODEL;
    const float* Yp = y + (size_t)b * D_MODEL * SEQ_L + l0;

    v8f acc = {0.f, 0.f, 0.f, 0.f, 0.f, 0.f, 0.f, 0.f};

    const int m  = lane & 15;
    const int kb = (lane < 16) ? 0 : 2;

#ifdef HAVE_TDM
    __shared__ float Wl[2][64][33];
    __shared__ float Yl[2][32][33];

    if (tid < 32) {                      // wave 0 drives the TDM
        tdm_load_2d((unsigned)(uintptr_t)&Wl[0][0][0], Wp,
                    32, 64, D_MODEL, D_MODEL, D_MODEL);
        tdm_load_2d((unsigned)(uintptr_t)&Yl[0][0][0], Yp,
                    32, 32, SEQ_L, D_MODEL, SEQ_L);
    }
    for (int s = 0; s < D_MODEL / 32; ++s) {
        if (tid < 32) {
            if (s + 1 < D_MODEL / 32) {  // prefetch next stage into other buffer
                const int nc = (s + 1) * 32;
                tdm_load_2d((unsigned)(uintptr_t)&Wl[(s + 1) & 1][0][0], Wp + nc,
                            32, 64, D_MODEL, D_MODEL, D_MODEL);
                tdm_load_2d((unsigned)(uintptr_t)&Yl[(s + 1) & 1][0][0],
                            Yp + (size_t)nc * SEQ_L,
                            32, 32, SEQ_L, D_MODEL, SEQ_L);
                __builtin_amdgcn_s_wait_tensorcnt(2);   // in-order: stage s landed
            } else {
                __builtin_amdgcn_s_wait_tensorcnt(0);
            }
        }
        __syncthreads();
        const float (*Wb)[33] = Wl[s & 1];
        const float (*Yb)[33] = Yl[s & 1];
        #pragma unroll
        for (int k = 0; k < 32; k += 4) {
            v2f a, bf;
            a.x  = Wb[oo + m][k + kb];
            a.y  = Wb[oo + m][k + kb + 1];
            bf.x = Yb[k + kb][ll + m];
            bf.y = Yb[k + kb + 1][ll + m];
            acc = __builtin_amdgcn_wmma_f32_16x16x4_f32(
                false, a, false, bf, (short)0, acc, false, false);
        }
        __syncthreads();
    }
#else
    __shared__ float Wl[64][33];
    __shared__ float Yl[32][33];

    for (int cc = 0; cc < D_MODEL; cc += 32) {
        #pragma unroll
        for (int i = 0; i < 8; ++i) {
            const int idx = tid + i * 256;
            const int r = idx >> 5, c = idx & 31;
            Wl[r][c] = Wp[(size_t)r * D_MODEL + cc + c];
        }
        #pragma unroll
        for (int i = 0; i < 4; ++i) {
            const int idx = tid + i * 256;
            const int r = idx >> 5, c = idx & 31;
            Yl[r][c] = Yp[(size_t)(cc + r) * SEQ_L + c];
        }
        const int nc = cc + 32;
        if (nc < D_MODEL) {
            const int pr = tid >> 2;
            const int pc = (tid & 3) * 8;
            __builtin_prefetch(&Wp[(size_t)pr * D_MODEL + nc + pc], 0, 1);
            if (tid < 128)
                __builtin_prefetch(&Yp[(size_t)(nc + pr) * SEQ_L + pc], 0, 1);
        }
        __syncthreads();
        #pragma unroll
        for (int k = 0; k < 32; k += 4) {
            v2f a, bf;
            a.x  = Wl[oo + m][k + kb];
            a.y  = Wl[oo + m][k + kb + 1];
            bf.x = Yl[k + kb][ll + m];
            bf.y = Yl[k + kb + 1][ll + m];
            acc = __builtin_amdgcn_wmma_f32_16x16x4_f32(
                false, a, false, bf, (short)0, acc, false, false);
        }
        __syncthreads();
    }
#endif

    const int ncol  = lane & 15;
    const int mbase = (lane < 16) ? 0 : 8;
    #pragma unroll
    for (int i = 0; i < 8; ++i) {
        const int o = o0 + oo + mbase + i;
        const int l = l0 + ll + ncol;
        const size_t oidx = ((size_t)b * D_MODEL + o) * SEQ_L + l;
        out[oidx] = acc[i] + b_out[o] + x[oidx];
    }
}

extern "C" void kernel_launch(void* const* d_in, const int* in_sizes, int n_in,
                              void* d_out, int out_size, void* d_ws, size_t ws_size,
                              hipStream_t stream) {
    const float* x          = (const float*)d_in[0];
    const float* log_A_real = (const float*)d_in[1];
    const float* A_imag     = (const float*)d_in[2];
    const float* B_real     = (const float*)d_in[3];
    const float* B_imag     = (const float*)d_in[4];
    const float* C_real     = (const float*)d_in[5];
    const float* C_imag     = (const float*)d_in[6];
    const float* Dv         = (const float*)d_in[7];
    const float* log_dt     = (const float*)d_in[8];
    const float* W_out      = (const float*)d_in[9];
    const float* b_out      = (const float*)d_in[10];
    float* out = (float*)d_out;

    // workspace layout (96 MiB total)
    float* y  = (float*)d_ws;                           // 8*512*2048
    float* Ur = y  + (size_t)BATCH * D_MODEL * SEQ_L;   // 512*64*256
    float* Ui = Ur + (size_t)D_MODEL * STATE_N * NCOL;  // 512*64*256

    // 1) U = P_d @ X_d  (complex chunk projection, WMMA)
    ssm_chunk_proj_kernel<<<dim3(NCOL / 64, D_MODEL), dim3(256), 0, stream>>>(
        x, log_A_real, A_imag, B_real, B_imag, log_dt, Ur, Ui);

    // 2) 32-step chunk-state scan (in-place U -> S_prev)
    ssm_chunk_scan_kernel<<<dim3((D_MODEL * STATE_N * BATCH) / 256), dim3(256), 0, stream>>>(
        log_A_real, A_imag, log_dt, Ur, Ui);

    // 3) y = L@X + Er@Sr - Ei@Si + D*x  (WMMA, 3 accumulate passes)
    ssm_chunk_out_kernel<<<dim3(NCOL / 64, D_MODEL), dim3(256), 0, stream>>>(
        x, log_A_real, A_imag, B_real, B_imag, C_real, C_imag, Dv, log_dt,
        Ur, Ui, y);

    // 4) out = W_out @ y + b_out + x  (WMMA GEMM, TDM double-buffered staging)
    outproj_wmma_kernel<<<dim3(SEQ_L / 32, D_MODEL / 64, BATCH), dim3(256), 0, stream>>>(
        y, x, W_out, b_out, out);
}